// Gnn_8177617732073
// MI455X (gfx1250) — compile-verified
//
#include <hip/hip_runtime.h>

#define N_NODES 100000
#define N_EDGES 1000000
#define D 64

typedef __attribute__((ext_vector_type(2))) float v2f;
typedef __attribute__((ext_vector_type(8))) float v8f;

// ---------------------------------------------------------------- zero buffer
__global__ void zero_f4(float4* __restrict__ p, int n4) {
    int i = blockIdx.x * blockDim.x + threadIdx.x;
    if (i < n4) {
        float4 z = {0.f, 0.f, 0.f, 0.f};
        p[i] = z;
    }
}

// ----------------------------------------------------- edge scatter (segment_sum)
// 64 threads per edge, one float atomic each: agg[dst[e]][c] += x[src[e]][c]
__global__ void scatter_add(const float* __restrict__ x,
                            const int* __restrict__ src,
                            const int* __restrict__ dst,
                            float* __restrict__ agg) {
    long long t = (long long)blockIdx.x * blockDim.x + threadIdx.x;
    int e = (int)(t >> 6);
    int c = (int)(t & 63);
    if (e < N_EDGES) {
        int s = src[e];
        int d = dst[e];
        unsafeAtomicAdd(&agg[(long long)d * D + c], x[(long long)s * D + c]);
    }
}

// --------------------------------------------------- fused GraphConv layer (WMMA)
// out[n][h] = tanh( sum_k agg[n][k]*Wrel[h][k] + brel[h] + sum_k x[n][k]*Wroot[h][k] )
// Block = 128 threads = 4 waves; block handles 16 node-rows, wave w handles
// output columns [16w, 16w+16).  V_WMMA_F32_16X16X4_F32, K=64 -> 16 steps/matrix.
__global__ void __launch_bounds__(128) fused_conv(
        const float* __restrict__ agg, const float* __restrict__ xin,
        const float* __restrict__ Wrel, const float* __restrict__ brel,
        const float* __restrict__ Wroot, float* __restrict__ hout) {
    const int lane = threadIdx.x & 31;
    const int wave = threadIdx.x >> 5;
    const int n    = lane & 15;   // A: row-in-tile; B/C/D: col-in-tile
    const int kh   = lane >> 4;   // which K-half (A/B) / which M-half (C/D)
    const int h0   = wave * 16;   // output column base
    const long long rowBase = (long long)blockIdx.x * 16;

    const float* aRow     = agg   + (rowBase + n) * D;   // A fragment source
    const float* xRow     = xin   + (rowBase + n) * D;
    const float* wRelRow  = Wrel  + (long long)(h0 + n) * D;  // B[k][n] = W[h0+n][k]
    const float* wRootRow = Wroot + (long long)(h0 + n) * D;

    v8f c;
    float bias = brel[h0 + n];
#pragma unroll
    for (int r = 0; r < 8; ++r) c[r] = bias;

    // agg @ Wrel^T
#pragma unroll
    for (int k0 = 0; k0 < D; k0 += 4) {
        v2f a = *(const v2f*)(aRow + k0 + 2 * kh);      // A: K=k0+2kh, k0+2kh+1
        v2f b = *(const v2f*)(wRelRow + k0 + 2 * kh);   // B: same K rows, col n
        c = __builtin_amdgcn_wmma_f32_16x16x4_f32(false, a, false, b,
                                                  (short)0, c, false, false);
    }
    // x @ Wroot^T (accumulate into same C)
#pragma unroll
    for (int k0 = 0; k0 < D; k0 += 4) {
        v2f a = *(const v2f*)(xRow + k0 + 2 * kh);
        v2f b = *(const v2f*)(wRootRow + k0 + 2 * kh);
        c = __builtin_amdgcn_wmma_f32_16x16x4_f32(false, a, false, b,
                                                  (short)0, c, false, false);
    }

    // tanh + store: VGPR r -> row (r + 8*kh), col (h0 + n)
#pragma unroll
    for (int r = 0; r < 8; ++r) {
        float v = tanhf(c[r]);
        hout[(rowBase + r + 8 * kh) * D + h0 + n] = v;
    }
}

// ------------------------------------------------------------- readout GEMV
// out[n] = dot(h[n][0:64], Wout[0:64]) + bout   (wave32: 2 elems/lane + shfl_xor)
__global__ void __launch_bounds__(256) readout(const float* __restrict__ h,
                                               const float* __restrict__ Wout,
                                               const float* __restrict__ bout,
                                               float* __restrict__ out) {
    int lane = threadIdx.x & 31;
    int wave = threadIdx.x >> 5;
    int node = blockIdx.x * 8 + wave;
    if (node >= N_NODES) return;
    const float* row = h + (long long)node * D;
    float s = row[lane] * Wout[lane] + row[lane + 32] * Wout[lane + 32];
#pragma unroll
    for (int off = 16; off; off >>= 1) s += __shfl_xor(s, off, 32);
    if (lane == 0) out[node] = s + bout[0];
}

// ---------------------------------------------------------------------------
extern "C" void kernel_launch(void* const* d_in, const int* in_sizes, int n_in,
                              void* d_out, int out_size, void* d_ws, size_t ws_size,
                              hipStream_t stream) {
    const float* x      = (const float*)d_in[0];
    const int*   ei     = (const int*)d_in[1];
    const int*   src    = ei;              // edge_index[0]
    const int*   dst    = ei + N_EDGES;    // edge_index[1]
    const float* Wrel0  = (const float*)d_in[2];
    const float* brel0  = (const float*)d_in[3];
    const float* Wroot0 = (const float*)d_in[4];
    const float* Wrel1  = (const float*)d_in[5];
    const float* brel1  = (const float*)d_in[6];
    const float* Wroot1 = (const float*)d_in[7];
    const float* Wout   = (const float*)d_in[8];
    const float* bout   = (const float*)d_in[9];
    float* out = (float*)d_out;

    const size_t featBytes = (size_t)N_NODES * D * sizeof(float);
    float* agg = (float*)d_ws;
    float* h1  = (float*)((char*)d_ws + featBytes);
    float* h2  = (float*)((char*)d_ws + 2 * featBytes);

    const int n4 = N_NODES * D / 4;
    dim3 zgrid((n4 + 255) / 256);
    dim3 sgrid((unsigned)(((long long)N_EDGES * 64) / 256));  // 250000
    dim3 ggrid(N_NODES / 16);                                 // 6250
    dim3 rgrid(N_NODES / 8);                                  // 12500

    // Layer 0
    zero_f4<<<zgrid, 256, 0, stream>>>((float4*)agg, n4);
    scatter_add<<<sgrid, 256, 0, stream>>>(x, src, dst, agg);
    fused_conv<<<ggrid, 128, 0, stream>>>(agg, x, Wrel0, brel0, Wroot0, h1);
    // Layer 1
    zero_f4<<<zgrid, 256, 0, stream>>>((float4*)agg, n4);
    scatter_add<<<sgrid, 256, 0, stream>>>(h1, src, dst, agg);
    fused_conv<<<ggrid, 128, 0, stream>>>(agg, h1, Wrel1, brel1, Wroot1, h2);
    // Readout
    readout<<<rgrid, 256, 0, stream>>>(h2, Wout, bout, out);
}